// ModelNew_70918499991668
// MI455X (gfx1250) — compile-verified
//
#include <hip/hip_runtime.h>

// CDNA5 / gfx1250 chunked gated-delta-rule (KDA) forward.
// fp32 WMMA 16x16x4 + TDM tensor_load_to_lds for the V-chunk staging copy.

typedef __attribute__((ext_vector_type(2))) float v2f;
typedef __attribute__((ext_vector_type(8))) float v8f;
typedef unsigned int u32x4 __attribute__((ext_vector_type(4)));
typedef int i32x8 __attribute__((ext_vector_type(8)));
typedef int i32x4 __attribute__((ext_vector_type(4)));

namespace {
constexpr int BDIM = 4, TDIM = 2048, HDIM = 16, KDIM = 128, VDIM = 128;
constexpr int TC  = 16;          // time chunk
constexpr int VS  = 32;          // V-slice handled per workgroup
constexpr int NTH = 128;         // 4 waves (wave32)
constexpr int KS  = 132;         // LDS stride (floats) for K-wide rows: 8B aligned, conflict-free
constexpr int VSS = 40;          // LDS stride for V-slice rows: 32 data + 8 pad dwords (TDM pads)
constexpr int MS  = 18;          // stride for 16x16 matrices (even -> b64-able A loads)
}

// A fragment: 16x4 fp32. lanes 0-15: M=lane, K={0,1}; lanes 16-31: M=lane-16, K={2,3}.
__device__ __forceinline__ v2f frag_a(const float* p, int rs, int r0, int c0, int lane) {
  const int m = lane & 15, kk = (lane >> 4) << 1;
  v2f a; a.x = p[(r0 + m) * rs + c0 + kk]; a.y = p[(r0 + m) * rs + c0 + kk + 1];
  return a;
}
// B fragment: 4x16 fp32, row (fixed K) striped across lanes within a VGPR.
__device__ __forceinline__ v2f frag_b(const float* p, int rs, int r0, int c0, int lane) {
  const int n = lane & 15, kk = (lane >> 4) << 1;
  v2f b; b.x = p[(r0 + kk) * rs + c0 + n]; b.y = p[(r0 + kk + 1) * rs + c0 + n];
  return b;
}
// B fragment from a transposed source: B[k][n] = p[n][k]
__device__ __forceinline__ v2f frag_bt(const float* p, int rs, int r0, int c0, int lane) {
  const int n = lane & 15, kk = (lane >> 4) << 1;
  v2f b; b.x = p[(c0 + n) * rs + r0 + kk]; b.y = p[(c0 + n) * rs + r0 + kk + 1];
  return b;
}

#define WMMA_F32(A, B, C) \
  __builtin_amdgcn_wmma_f32_16x16x4_f32(false, (A), false, (B), (short)0, (C), false, false)

__global__ __launch_bounds__(NTH)
void kda_chunked_kernel(const float* __restrict__ q, const float* __restrict__ k,
                        const float* __restrict__ v, const float* __restrict__ g,
                        const float* __restrict__ beta, float* __restrict__ out) {
  __shared__ float sS[KDIM][VSS];   // state slice S[K, VS]
  __shared__ float sW[TC][KS];      // k * exp(G)
  __shared__ float sU[TC][KS];      // k * exp(-G)
  __shared__ float sQ[TC][KS];      // q * exp(G) * scale
  __shared__ float sD[TC][VSS];     // v chunk (TDM) -> rhs -> delta
  __shared__ float sM[TC][MS];      // W @ U^T   (delta-solve matrix)
  __shared__ float sAm[TC][MS];     // tril(Q~ @ U^T) * diag(beta)
  __shared__ float sBeta[TC];
  __shared__ float sDend[KDIM];     // exp(G_15): end-of-chunk decay per key channel

  const int tid  = threadIdx.x;
  const int lane = tid & 31;
  const int wv   = tid >> 5;
  const int nvs  = VDIM / VS;                 // 4 V-slices
  const int vsl  = blockIdx.x % nvs;
  const int bh   = blockIdx.x / nvs;
  const int h    = bh % HDIM;
  const int b    = bh / HDIM;
  const int v0   = vsl * VS;
  const float scale = 0.088388347648318447f;  // 1/sqrt(128)

  const size_t rowstr   = (size_t)HDIM * KDIM;   // t-stride, q/k/g
  const size_t vrstr    = (size_t)HDIM * VDIM;   // t-stride, v/out
  const size_t base_qkg = (size_t)b * TDIM * HDIM * KDIM + (size_t)h * KDIM;
  const size_t base_v   = (size_t)b * TDIM * HDIM * VDIM + (size_t)h * VDIM + v0;

  for (int i = tid; i < KDIM * VSS; i += NTH) sS[i / VSS][i % VSS] = 0.0f;
  __syncthreads();

  for (int t0 = 0; t0 < TDIM; t0 += TC) {
    v8f oacc0 = {0.f, 0.f, 0.f, 0.f, 0.f, 0.f, 0.f, 0.f};  // wave1: O partial, tile 0
    v8f oacc1 = {0.f, 0.f, 0.f, 0.f, 0.f, 0.f, 0.f, 0.f};  // wave1: O partial, tile 1

    // ---- Phase A: TDM v-chunk -> LDS (wave 3) + per-channel decay preprocessing ----
    if (wv == 3) {
      // Tensor DMA descriptor: 2D tile [16 rows x 32 fp32], row stride H*V elements,
      // LDS padding 8 dwords per 32 dwords -> lands directly in sD[TC][VSS].
      const unsigned lds_addr = (unsigned)(size_t)&sD[0][0];
      const unsigned long long ga =
          (unsigned long long)(size_t)(v + base_v + (size_t)t0 * vrstr);
      u32x4 g0;
      g0.x = 1u;                                   // count=1 valid descriptor
      g0.y = lds_addr;                             // lds_addr
      g0.z = (unsigned)ga;                         // global_addr[31:0]
      g0.w = (unsigned)(ga >> 32) | 0x80000000u;   // global_addr[56:32] | type=2
      i32x8 g1;
      g1[0] = 0x0F120000;            // data_size=4B | pad_enable | interval=32dw | amount=8dw
      g1[1] = (int)(32u << 16);      // tensor_dim0 = 32
      g1[2] = (int)(16u << 16);      // tensor_dim1 = 16
      g1[3] = (int)(32u << 16);      // tile_dim0 = 32
      g1[4] = 16;                    // tile_dim1 = 16, tile_dim2 = 0
      g1[5] = (int)(HDIM * VDIM);    // tensor_dim0_stride = 2048 elements
      g1[6] = 0; g1[7] = 0;
      const i32x4 z4 = {0, 0, 0, 0};
      const i32x8 z8 = {0, 0, 0, 0, 0, 0, 0, 0};
      __builtin_amdgcn_tensor_load_to_lds(g0, g1, z4, z4, z8, 0);
    }
    {
      const int ch = tid;  // NTH == KDIM
      const float* gp = g + base_qkg + (size_t)t0 * rowstr + ch;
      const float* kp = k + base_qkg + (size_t)t0 * rowstr + ch;
      const float* qp = q + base_qkg + (size_t)t0 * rowstr + ch;
      float G = 0.0f, e = 1.0f;
#pragma unroll
      for (int t = 0; t < TC; ++t) {
        G += gp[(size_t)t * rowstr];
        e = __expf(G);
        const float kv = kp[(size_t)t * rowstr];
        sW[t][ch] = kv * e;
        sU[t][ch] = kv * __expf(-G);
        sQ[t][ch] = qp[(size_t)t * rowstr] * e * scale;
      }
      sDend[ch] = e;
      if (tid < TC) sBeta[tid] = beta[((size_t)b * TDIM + t0 + tid) * HDIM + h];
    }
    if (wv == 3) __builtin_amdgcn_s_wait_tensorcnt((short)0);
    __syncthreads();

    // ---- Phase B: wave0 rhs = V - W@S0 (2 tiles, shared A) ; wave1 O = Q~@S0 (regs) ;
    //               wave2 M = W@U^T ; wave3 Aq = Q~@U^T ----
    if (wv == 0) {
      v8f a0 = {0.f, 0.f, 0.f, 0.f, 0.f, 0.f, 0.f, 0.f};
      v8f a1 = {0.f, 0.f, 0.f, 0.f, 0.f, 0.f, 0.f, 0.f};
#pragma unroll
      for (int kk = 0; kk < KDIM; kk += 4) {
        v2f a  = frag_a(&sW[0][0], KS, 0, kk, lane);
        v2f b0 = frag_b(&sS[0][0], VSS, kk, 0, lane);
        v2f b1 = frag_b(&sS[0][0], VSS, kk, 16, lane);
        a0 = WMMA_F32(a, b0, a0);
        a1 = WMMA_F32(a, b1, a1);
      }
      const int cm = (lane >> 4) << 3, cn = lane & 15;
#pragma unroll
      for (int j = 0; j < 8; ++j) {
        sD[j + cm][cn]      -= a0[j];
        sD[j + cm][16 + cn] -= a1[j];
      }
    } else if (wv == 1) {
#pragma unroll
      for (int kk = 0; kk < KDIM; kk += 4) {
        v2f a  = frag_a(&sQ[0][0], KS, 0, kk, lane);
        v2f b0 = frag_b(&sS[0][0], VSS, kk, 0, lane);
        v2f b1 = frag_b(&sS[0][0], VSS, kk, 16, lane);
        oacc0 = WMMA_F32(a, b0, oacc0);
        oacc1 = WMMA_F32(a, b1, oacc1);
      }
    } else {
      const float* A = (wv == 2) ? &sW[0][0] : &sQ[0][0];
      float* dst     = (wv == 2) ? &sM[0][0] : &sAm[0][0];
      v8f acc = {0.f, 0.f, 0.f, 0.f, 0.f, 0.f, 0.f, 0.f};
#pragma unroll
      for (int kk = 0; kk < KDIM; kk += 4) {
        v2f a  = frag_a(A, KS, 0, kk, lane);
        v2f bb = frag_bt(&sU[0][0], KS, kk, 0, lane);
        acc = WMMA_F32(a, bb, acc);
      }
      const int cm = (lane >> 4) << 3, cn = lane & 15;
#pragma unroll
      for (int j = 0; j < 8; ++j) dst[(j + cm) * MS + cn] = acc[j];
    }
    __syncthreads();

    // ---- Phase C: unit-lower-triangular solve (per V-column, wave 0) + mask Aq ----
    if (tid < VS) {
      const int c = tid;
#pragma unroll
      for (int t = 1; t < TC; ++t) {
        float a2 = sD[t][c];
        for (int s = 0; s < t; ++s) a2 -= sM[t][s] * sBeta[s] * sD[s][c];
        sD[t][c] = a2;
      }
    } else if (wv >= 2) {
      for (int i = tid - 2 * 32; i < TC * TC; i += 2 * 32) {
        const int t = i >> 4, s = i & 15;
        sAm[t][s] = (s <= t) ? sAm[t][s] * sBeta[s] : 0.0f;
      }
    }
    __syncthreads();

    // ---- Phase D: wave1 O += Am@Delta, store to global; waves 2,3 prefetch ----
    if (wv == 1) {
#pragma unroll
      for (int kk = 0; kk < TC; kk += 4) {
        v2f a  = frag_a(&sAm[0][0], MS, 0, kk, lane);
        v2f b0 = frag_b(&sD[0][0], VSS, kk, 0, lane);
        v2f b1 = frag_b(&sD[0][0], VSS, kk, 16, lane);
        oacc0 = WMMA_F32(a, b0, oacc0);
        oacc1 = WMMA_F32(a, b1, oacc1);
      }
      const int cm = (lane >> 4) << 3, cn = lane & 15;
#pragma unroll
      for (int j = 0; j < 8; ++j) {
        const size_t r = base_v + (size_t)(t0 + j + cm) * vrstr;
        out[r + cn]      = oacc0[j];
        out[r + 16 + cn] = oacc1[j];
      }
    } else if (wv >= 2 && t0 + TC < TDIM) {
      const int t = lane & 15;
      const float* p;
      if (wv == 2)
        p = ((lane < 16) ? q : k) + base_qkg + (size_t)(t0 + TC + t) * rowstr;
      else
        p = (lane < 16) ? (g + base_qkg + (size_t)(t0 + TC + t) * rowstr)
                        : (v + base_v + (size_t)(t0 + TC + t) * vrstr);
      __builtin_prefetch(p, 0, 3);
      __builtin_prefetch(p + 64, 0, 3);
    }
    __syncthreads();

    // ---- Phase E: S <- diag(exp(G15)) * (S0 + U^T diag(beta) Delta), all 4 waves ----
#pragma unroll
    for (int st = wv * 2; st < wv * 2 + 2; ++st) {
      const int k0 = st * 16;
      const int m = lane & 15;
      v2f af[TC / 4];                 // A fragments hoisted: reused for both V tiles
#pragma unroll
      for (int kq = 0; kq < TC / 4; ++kq) {
        const int s0 = kq * 4 + ((lane >> 4) << 1);
        af[kq].x = sBeta[s0] * sU[s0][k0 + m];
        af[kq].y = sBeta[s0 + 1] * sU[s0 + 1][k0 + m];
      }
      const int cm = (lane >> 4) << 3, cn = lane & 15;
#pragma unroll
      for (int vb = 0; vb < 2; ++vb) {
        v8f acc;
#pragma unroll
        for (int j = 0; j < 8; ++j) acc[j] = sS[k0 + j + cm][vb * 16 + cn];
#pragma unroll
        for (int kq = 0; kq < TC / 4; ++kq) {
          v2f bb = frag_b(&sD[0][0], VSS, kq * 4, vb * 16, lane);
          acc = WMMA_F32(af[kq], bb, acc);
        }
#pragma unroll
        for (int j = 0; j < 8; ++j) {
          const int r = k0 + j + cm;
          sS[r][vb * 16 + cn] = acc[j] * sDend[r];
        }
      }
    }
    __syncthreads();
  }
}

extern "C" void kernel_launch(void* const* d_in, const int* in_sizes, int n_in,
                              void* d_out, int out_size, void* d_ws, size_t ws_size,
                              hipStream_t stream) {
  (void)in_sizes; (void)n_in; (void)out_size; (void)d_ws; (void)ws_size;
  const float* q    = (const float*)d_in[0];
  const float* k    = (const float*)d_in[1];
  const float* v    = (const float*)d_in[2];
  const float* g    = (const float*)d_in[3];
  const float* beta = (const float*)d_in[4];
  float* out = (float*)d_out;
  dim3 grid(BDIM * HDIM * (VDIM / VS));   // 256 workgroups
  kda_chunked_kernel<<<grid, NTH, 0, stream>>>(q, k, v, g, beta, out);
}